// MultiBoxLoss_67499706024304
// MI455X (gfx1250) — compile-verified
//
#include <hip/hip_runtime.h>
#include <math.h>

#define TPB 256
#define GMAX 64   // max truths per image supported for LDS staging

typedef float v2f __attribute__((ext_vector_type(2)));
typedef float v8f __attribute__((ext_vector_type(8)));

__device__ __forceinline__ float sl1(float x) {
    float a = fabsf(x);
    return (a < 1.0f) ? 0.5f * a * a : a - 0.5f;
}

// CDNA5 async global->LDS copy (ASYNCcnt path), per CDNA5 ISA.
// GVS mode: copies 4 bytes from the SGPR-pair base plus the VGPR byte offset to the LDS address in %0.
__device__ __forceinline__ void async_load_f32_to_lds(uint32_t lds_off,
                                                      uint32_t byte_off,
                                                      const void* sbase) {
    asm volatile("global_load_async_to_lds_b32 %0, %1, %2"
                 :
                 : "v"(lds_off), "v"(byte_off), "s"(sbase)
                 : "memory");
}
__device__ __forceinline__ void wait_asynccnt0() {
    asm volatile("s_wait_asynccnt 0x0" ::: "memory");
}

// ---------------------------------------------------------------------------
// Kernel 0: zero integer counters and (padded) partial-sum arrays
// ---------------------------------------------------------------------------
__global__ void init_kernel(int* num_pos, int* tot, int B,
                            float* pl, float* plm, float* pcp, float* pcn,
                            int nbpad) {
    int t = blockIdx.x * blockDim.x + threadIdx.x;
    if (t < B) num_pos[t] = 0;
    if (t < 2) tot[t] = 0;
    if (t < nbpad) { pl[t] = 0.0f; plm[t] = 0.0f; pcp[t] = 0.0f; pcn[t] = 0.0f; }
}

// ---------------------------------------------------------------------------
// Kernel 1: per (image b, truth g): best prior (max IoU + first argmax over P)
// ---------------------------------------------------------------------------
__global__ void best_prior_kernel(const float* __restrict__ targets,
                                  const float* __restrict__ priors,
                                  int P, int G,
                                  int* __restrict__ bp_idx,
                                  int* __restrict__ bp_valid,
                                  int* __restrict__ anyv) {
    int b = blockIdx.x;
    int t = threadIdx.x;
    __shared__ float sv[TPB];
    __shared__ int   si[TPB];
    int anyflag = 0;  // meaningful in thread 0 only

    for (int g = 0; g < G; ++g) {
        const float* tr = targets + ((size_t)b * G + g) * 15;
        float ax1 = tr[0], ay1 = tr[1], ax2 = tr[2], ay2 = tr[3];
        float areaA = (ax2 - ax1) * (ay2 - ay1);

        float best = -1.0f;
        int   bidx = 0;
        for (int i = t; i < P; i += TPB) {
            const float* pr = priors + (size_t)i * 4;
            float cx = pr[0], cy = pr[1], pw = pr[2], ph = pr[3];
            float bx1 = cx - 0.5f * pw, by1 = cy - 0.5f * ph;
            float bx2 = cx + 0.5f * pw, by2 = cy + 0.5f * ph;
            float ltx = fmaxf(ax1, bx1), lty = fmaxf(ay1, by1);
            float rbx = fminf(ax2, bx2), rby = fminf(ay2, by2);
            float w = fmaxf(rbx - ltx, 0.0f), h = fmaxf(rby - lty, 0.0f);
            float inter = w * h;
            float areaB = (bx2 - bx1) * (by2 - by1);
            float iou = inter / (areaA + areaB - inter);
            if (iou > best) { best = iou; bidx = i; }   // strict > : first max
        }
        sv[t] = best; si[t] = bidx;
        __syncthreads();
        for (int s = TPB / 2; s > 0; s >>= 1) {
            if (t < s) {
                float v2 = sv[t + s]; int i2 = si[t + s];
                if (v2 > sv[t] || (v2 == sv[t] && i2 < si[t])) { sv[t] = v2; si[t] = i2; }
            }
            __syncthreads();
        }
        if (t == 0) {
            int valid = (sv[0] >= 0.2f) ? 1 : 0;
            bp_idx[b * G + g]   = si[0];
            bp_valid[b * G + g] = valid;
            if (valid) anyflag = 1;
        }
        __syncthreads();
    }
    if (t == 0) anyv[b] = anyflag;
}

// ---------------------------------------------------------------------------
// Kernel 2: per prior: best truth + forced overrides, conf, encode, losses.
// Targets row and best-prior tables staged in LDS once per block.
// ---------------------------------------------------------------------------
__global__ void match_kernel(const float* __restrict__ cls,
                             const float* __restrict__ locd,
                             const float* __restrict__ lmd,
                             const float* __restrict__ priors,
                             const float* __restrict__ targets,
                             int P, int G, int C, int CPB,
                             const int* __restrict__ bp_idx,
                             const int* __restrict__ bp_valid,
                             const int* __restrict__ anyv,
                             int* __restrict__ num_pos,
                             int* __restrict__ tot,
                             float* __restrict__ lossc,
                             int* __restrict__ posArr,
                             float* __restrict__ pl,
                             float* __restrict__ plm,
                             float* __restrict__ pcp) {
    int b = blockIdx.x / CPB;
    int i = (blockIdx.x % CPB) * TPB + threadIdx.x;
    int t = threadIdx.x;

    __shared__ float st[GMAX * 15];
    __shared__ int   sbp[GMAX];
    __shared__ int   sbv[GMAX];
    for (int k = t; k < G * 15; k += TPB) st[k] = targets[(size_t)b * G * 15 + k];
    if (t < G) { sbp[t] = bp_idx[b * G + t]; sbv[t] = bp_valid[b * G + t]; }
    __syncthreads();

    float ll = 0.0f, llm = 0.0f, cpos = 0.0f;
    int cp = 0, cp1 = 0;

    if (i < P) {
        const float* pr = priors + (size_t)i * 4;
        float pcx = pr[0], pcy = pr[1], pw = pr[2], ph = pr[3];
        float bx1 = pcx - 0.5f * pw, by1 = pcy - 0.5f * ph;
        float bx2 = pcx + 0.5f * pw, by2 = pcy + 0.5f * ph;
        float areaB = (bx2 - bx1) * (by2 - by1);

        // best truth (max over g, first argmax)
        float best = -1.0f; int bidx = 0;
        for (int g = 0; g < G; ++g) {
            const float* tr = st + g * 15;
            float ax1 = tr[0], ay1 = tr[1], ax2 = tr[2], ay2 = tr[3];
            float ltx = fmaxf(ax1, bx1), lty = fmaxf(ay1, by1);
            float rbx = fminf(ax2, bx2), rby = fminf(ay2, by2);
            float w = fmaxf(rbx - ltx, 0.0f), h = fmaxf(rby - lty, 0.0f);
            float inter = w * h;
            float areaA = (ax2 - ax1) * (ay2 - ay1);
            float iou = inter / (areaA + areaB - inter);
            if (iou > best) { best = iou; bidx = g; }
        }
        // forced assignment: scatter-max emulated by gather over G
        int fidx = -1; float fov = 0.0f;
        for (int g = 0; g < G; ++g) {
            if (sbp[g] == i) {
                if (g > fidx) fidx = g;
                if (sbv[g]) fov = 2.0f;
            }
        }
        float btov = fmaxf(best, fov);
        int bt = (fidx >= 0) ? fidx : bidx;

        const float* tr = st + bt * 15;
        int conf = (int)tr[14];           // labels are -1.0 / 1.0
        if (btov < 0.35f) conf = 0;
        if (!anyv[b])     conf = 0;
        int pos  = (conf != 0);
        int pos1 = (conf > 0);

        if (pos) {
            float mx1 = tr[0], my1 = tr[1], mx2 = tr[2], my2 = tr[3];
            float gx = ((mx1 + mx2) * 0.5f - pcx) / (0.1f * pw);
            float gy = ((my1 + my2) * 0.5f - pcy) / (0.1f * ph);
            float gw = logf((mx2 - mx1) / pw) / 0.2f;
            float gh = logf((my2 - my1) / ph) / 0.2f;
            const float* ld = locd + ((size_t)b * P + i) * 4;
            ll = sl1(ld[0] - gx) + sl1(ld[1] - gy) + sl1(ld[2] - gw) + sl1(ld[3] - gh);
        }
        if (pos1) {
            const float* lm = lmd + ((size_t)b * P + i) * 10;
            for (int k = 0; k < 5; ++k) {
                float gx = (tr[4 + 2 * k] - pcx) / (0.1f * pw);
                float gy = (tr[5 + 2 * k] - pcy) / (0.1f * ph);
                llm += sl1(lm[2 * k] - gx) + sl1(lm[2 * k + 1] - gy);
            }
        }
        // stable logsumexp - gold  (conf_t -> 1 for pos, 0 otherwise)
        const float* x = cls + ((size_t)b * P + i) * C;
        float m = x[0];
        for (int c = 1; c < C; ++c) m = fmaxf(m, x[c]);
        float s = 0.0f;
        for (int c = 0; c < C; ++c) s += expf(x[c] - m);
        float logz = m + logf(s);
        float lg = logz - x[pos ? 1 : 0];

        lossc[(size_t)b * P + i]  = pos ? 0.0f : lg;
        posArr[(size_t)b * P + i] = pos;
        cpos = pos ? lg : 0.0f;
        cp = pos; cp1 = pos1;
    }

    // block reductions
    __shared__ float sf[TPB];
    __shared__ int   si2[TPB];

    sf[t] = ll; __syncthreads();
    for (int s = TPB / 2; s > 0; s >>= 1) { if (t < s) sf[t] += sf[t + s]; __syncthreads(); }
    if (t == 0) pl[blockIdx.x] = sf[0];
    __syncthreads();

    sf[t] = llm; __syncthreads();
    for (int s = TPB / 2; s > 0; s >>= 1) { if (t < s) sf[t] += sf[t + s]; __syncthreads(); }
    if (t == 0) plm[blockIdx.x] = sf[0];
    __syncthreads();

    sf[t] = cpos; __syncthreads();
    for (int s = TPB / 2; s > 0; s >>= 1) { if (t < s) sf[t] += sf[t + s]; __syncthreads(); }
    if (t == 0) pcp[blockIdx.x] = sf[0];
    __syncthreads();

    si2[t] = cp; __syncthreads();
    for (int s = TPB / 2; s > 0; s >>= 1) { if (t < s) si2[t] += si2[t + s]; __syncthreads(); }
    if (t == 0) { atomicAdd(&num_pos[b], si2[0]); atomicAdd(&tot[0], si2[0]); }
    __syncthreads();

    si2[t] = cp1; __syncthreads();
    for (int s = TPB / 2; s > 0; s >>= 1) { if (t < s) si2[t] += si2[t + s]; __syncthreads(); }
    if (t == 0) atomicAdd(&tot[1], si2[0]);
}

// ---------------------------------------------------------------------------
// Kernel 3: hard-negative mining by rank counting (stable descending order):
// rank_i = #{j : lc[j] > lc[i]}  +  #{j < i : lc[j] == lc[i]}
// Tile fill via async global->LDS (ASYNCcnt); inner loop reads ds_load_b128.
// ---------------------------------------------------------------------------
__global__ void neg_kernel(const float* __restrict__ lossc,
                           const int* __restrict__ posArr,
                           const int* __restrict__ num_pos,
                           int P, int CPB, int negpos,
                           float* __restrict__ pcn) {
    int b = blockIdx.x / CPB;
    int i = (blockIdx.x % CPB) * TPB + threadIdx.x;
    int t = threadIdx.x;

    float lci = 0.0f; int posi = 1;
    if (i < P) {
        lci  = lossc[(size_t)b * P + i];
        posi = posArr[(size_t)b * P + i];
    }
    int nn = negpos * num_pos[b]; if (nn > P - 1) nn = P - 1;

    const float* row = lossc + (size_t)b * P;
    __shared__ __align__(16) float tile[TPB];
    uint32_t lds_off = (uint32_t)(uintptr_t)(&tile[t]);

    int rank = 0;
    for (int base = 0; base < P; base += TPB) {
        int j = base + t; if (j > P - 1) j = P - 1;   // clamp: no EXEC divergence
        async_load_f32_to_lds(lds_off, (uint32_t)j * 4u, row);
        wait_asynccnt0();
        __syncthreads();

        int lim = P - base; if (lim > TPB) lim = TPB;
        int ng4 = lim >> 2;
        if (i < P) {
            const float4* t4 = (const float4*)tile;
            for (int k4 = 0; k4 < ng4; ++k4) {
                float4 v = t4[k4];
                int jj = base + (k4 << 2);
                rank += (v.x > lci) + ((v.x == lci) & (jj + 0 < i));
                rank += (v.y > lci) + ((v.y == lci) & (jj + 1 < i));
                rank += (v.z > lci) + ((v.z == lci) & (jj + 2 < i));
                rank += (v.w > lci) + ((v.w == lci) & (jj + 3 < i));
            }
            for (int k = ng4 << 2; k < lim; ++k) {
                float v = tile[k];
                rank += (v > lci) + ((v == lci) & (base + k < i));
            }
        }
        __syncthreads();
    }

    float contrib = (i < P && !posi && rank < nn) ? lci : 0.0f;

    __shared__ float sf[TPB];
    sf[t] = contrib; __syncthreads();
    for (int s = TPB / 2; s > 0; s >>= 1) { if (t < s) sf[t] += sf[t + s]; __syncthreads(); }
    if (t == 0) pcn[blockIdx.x] = sf[0];
}

// ---------------------------------------------------------------------------
// Kernel 4: single-wave final reduction using V_WMMA_F32_16X16X4_F32.
// n is padded to a multiple of 64 and the pad is pre-zeroed -> branch-free:
// one global_load_b64 + one wmma per 64 elements. Exact fp32 on matrix pipe.
// ---------------------------------------------------------------------------
__device__ float wmma_reduce(const float* __restrict__ buf, int n64) {
    int lane = threadIdx.x;  // 0..31, EXEC all ones (required for WMMA)
    v8f c = {0.f, 0.f, 0.f, 0.f, 0.f, 0.f, 0.f, 0.f};
    v2f ones = {1.0f, 1.0f};
    const v2f* b2 = (const v2f*)buf;
    for (int base = 0; base < n64; base += 64) {
        v2f a = b2[(base >> 1) + lane];   // elements base+2*lane, base+2*lane+1
        // 8 args: (neg_a, A, neg_b, B, c_mod, C, reuse_a, reuse_b)
        c = __builtin_amdgcn_wmma_f32_16x16x4_f32(false, a, false, ones,
                                                  (short)0, c, false, false);
    }
    // C[m][n] identical over n. Lane L: c[r] = C[r + (L>=16 ? 8:0)][L%16]
    float tsum = c[0] + c[1] + c[2] + c[3] + c[4] + c[5] + c[6] + c[7];
    __shared__ float sh[32];
    sh[lane] = tsum;
    __syncthreads();
    float r = sh[0] + sh[16];  // rows 0..7 + rows 8..15 (column 0)
    __syncthreads();
    return r;
}

__global__ void final_kernel(const float* __restrict__ pl,
                             const float* __restrict__ plm,
                             const float* __restrict__ pcp,
                             const float* __restrict__ pcn,
                             int nbpad, const int* __restrict__ tot,
                             float* __restrict__ out) {
    float sl  = wmma_reduce(pl,  nbpad);
    float slm = wmma_reduce(plm, nbpad);
    float sp  = wmma_reduce(pcp, nbpad);
    float sn  = wmma_reduce(pcn, nbpad);
    if (threadIdx.x == 0) {
        float N  = fmaxf((float)tot[0], 1.0f);
        float N1 = fmaxf((float)tot[1], 1.0f);
        out[0] = sl / N;
        out[1] = (sp + sn) / N;
        out[2] = slm / N1;
    }
}

// ---------------------------------------------------------------------------
extern "C" void kernel_launch(void* const* d_in, const int* in_sizes, int n_in,
                              void* d_out, int out_size, void* d_ws, size_t ws_size,
                              hipStream_t stream) {
    const float* cls     = (const float*)d_in[0];
    const float* locd    = (const float*)d_in[1];
    const float* lmd     = (const float*)d_in[2];
    const float* priors  = (const float*)d_in[3];
    const float* targets = (const float*)d_in[4];

    int P = in_sizes[3] / 4;
    int B = in_sizes[1] / (P * 4);
    int C = in_sizes[0] / (B * P);
    int G = in_sizes[4] / (B * 15);
    if (G > GMAX) G = GMAX;
    int CPB = (P + TPB - 1) / TPB;
    int NB = B * CPB;
    int NBpad = (NB + 63) & ~63;          // pad to WMMA chunk (64 f32)

    // bump-allocate workspace (256B aligned)
    char* w = (char*)d_ws;
    auto alloc = [&](size_t bytes) {
        void* p = (void*)w;
        w += (bytes + 255) & ~(size_t)255;
        return p;
    };
    int*   bp_idx   = (int*)alloc((size_t)B * G * sizeof(int));
    int*   bp_valid = (int*)alloc((size_t)B * G * sizeof(int));
    int*   anyv     = (int*)alloc((size_t)B * sizeof(int));
    int*   num_pos  = (int*)alloc((size_t)B * sizeof(int));
    int*   tot      = (int*)alloc(2 * sizeof(int));
    float* lossc    = (float*)alloc((size_t)B * P * sizeof(float));
    int*   posArr   = (int*)alloc((size_t)B * P * sizeof(int));
    float* pl       = (float*)alloc((size_t)NBpad * sizeof(float));
    float* plm      = (float*)alloc((size_t)NBpad * sizeof(float));
    float* pcp      = (float*)alloc((size_t)NBpad * sizeof(float));
    float* pcn      = (float*)alloc((size_t)NBpad * sizeof(float));

    int initN = NBpad > B ? NBpad : B;
    init_kernel<<<(initN + TPB - 1) / TPB, TPB, 0, stream>>>(num_pos, tot, B,
                                                             pl, plm, pcp, pcn, NBpad);
    best_prior_kernel<<<B, TPB, 0, stream>>>(targets, priors, P, G,
                                             bp_idx, bp_valid, anyv);
    match_kernel<<<NB, TPB, 0, stream>>>(cls, locd, lmd, priors, targets,
                                         P, G, C, CPB,
                                         bp_idx, bp_valid, anyv,
                                         num_pos, tot, lossc, posArr,
                                         pl, plm, pcp);
    neg_kernel<<<NB, TPB, 0, stream>>>(lossc, posArr, num_pos, P, CPB, 7, pcn);
    final_kernel<<<1, 32, 0, stream>>>(pl, plm, pcp, pcn, NBpad, tot, (float*)d_out);
}